// GCN_1_18056042512527
// MI455X (gfx1250) — compile-verified
//
#include <hip/hip_runtime.h>
#include <hip/hip_bf16.h>

typedef __attribute__((ext_vector_type(2))) float v2f;
typedef __attribute__((ext_vector_type(4))) float vf4;
typedef __attribute__((ext_vector_type(8))) float v8f;

#define NB    32
#define NT    3
#define NN    512
#define DD    128
#define NG    (NB*NT)          // 96 batched (b,t) slices

// ---------------- row sums -> d^{-1/2} (includes +1 from self loop) ------
__global__ __launch_bounds__(256) void rowsum_kernel(const float* __restrict__ E,
                                                     float* __restrict__ dinv) {
    int wave = threadIdx.x >> 5;
    int lane = threadIdx.x & 31;
    long row = (long)blockIdx.x * 8 + wave;      // over NG*NN rows
    const float* p = E + row * NN;
    float s = 0.f;
    #pragma unroll
    for (int m = 0; m < NN / 32; ++m) s += p[lane + 32 * m];
    #pragma unroll
    for (int off = 16; off > 0; off >>= 1) s += __shfl_down(s, off, 32);
    if (lane == 0) dinv[row] = rsqrtf(s + 1.0f);
}

// ---------------- En = dinv_i * (E + I) * dinv_j -------------------------
__global__ __launch_bounds__(256) void en_kernel(const float* __restrict__ E,
                                                 const float* __restrict__ dinv,
                                                 float* __restrict__ En) {
    long f = (long)blockIdx.x * blockDim.x + threadIdx.x;  // float4 index
    int j4 = (int)(f & 127);            // 128 float4 per row of 512
    int i  = (int)((f >> 7) & 511);
    int bt = (int)(f >> 16);            // / (128*512)
    long base = (((long)bt * NN + i) * NN) + (long)j4 * 4;
    float di = dinv[bt * NN + i];
    const float* dj = dinv + bt * NN + j4 * 4;
    vf4 e = *(const vf4*)(E + base);
    vf4 o;
    #pragma unroll
    for (int q = 0; q < 4; ++q) {
        float v = e[q];
        if (j4 * 4 + q == i) v += 1.0f;
        o[q] = di * v * dj[q];
    }
    *(vf4*)(En + base) = o;
}

// ---------------- H0[g=b*T+t][i][:] = embed[N[b][i]] ---------------------
__global__ __launch_bounds__(256) void gather_kernel(const int* __restrict__ N,
                                                     const float* __restrict__ embed,
                                                     float* __restrict__ H) {
    long f = (long)blockIdx.x * blockDim.x + threadIdx.x;  // float4 over [NG,NN,DD]
    int d4 = (int)(f & 31);             // 32 float4 per D row
    int i  = (int)((f >> 5) & 511);
    int g  = (int)(f >> 14);            // / (32*512)
    int b  = g / NT;
    int node = N[b * NN + i];
    vf4 v = *(const vf4*)(embed + (long)node * DD + (long)d4 * 4);
    *(vf4*)(H + f * 4) = v;
}

// ---------------- batched GEMM via V_WMMA_F32_16X16X4_F32 ----------------
// C[g] (512 x 128) = A[g] (512 x K, row-major, lda=K) @ B[bi] (K x 128)
// bi = bMod ? g % bMod : g.  Optional relu on output.
//
// Block = 256 threads = 8 waves; 128x128 output tile.
// Wave tile = 32(M) x 64(N): wave w -> rows [32*(w&3), +32), cols [64*(w>>2), +64).
// Each B fragment feeds 2 WMMAs (rows m and m+16) -> ~33% less LDS read traffic.
//
// LDS layouts:
//   As[m][c]        row stride 36  -> A-frag reads conflict-free
//   Bs[k/2][2n+k&1] pair-row stride 288 (288%64==32) -> each B fragment is
//   one aligned ds_load_b64 of the (k,k+1) pair; lane halves hit disjoint
//   32-bank sets -> conflict-free, no register re-pairing movs.
#define KC    32
#define ALDS  (KC + 4)     // 36
#define PSTR  288          // floats per interleaved k-pair row of B

__global__ __launch_bounds__(256) void gemm_wmma(
    const float* __restrict__ A, long aStride, int K,
    const float* __restrict__ Bm, long bStride, int bMod,
    float* __restrict__ C, int relu)
{
    __shared__ float As[128 * ALDS];        // 18432 B
    __shared__ float Bs[(KC / 2) * PSTR];   // 18432 B

    int g  = blockIdx.y;
    int mt = blockIdx.x;                                  // 128-row tile
    const float* Ab = A + (long)g * aStride + (long)mt * 128 * K;
    int bi = bMod ? (g % bMod) : g;
    const float* Bb = Bm + (long)bi * bStride;
    float* Cb = C + ((long)g * NN + (long)mt * 128) * DD;

    int tid  = threadIdx.x;
    int lane = tid & 31;
    int wave = tid >> 5;
    int mrow = lane & 15;          // M for A-frag / N for B-frag
    int kh   = (lane >> 4) * 2;    // K offset of this lane half
    int m0   = (wave & 3) * 32;    // wave's row base (2 16-row tiles)
    int nh   = (wave >> 2) * 64;   // wave's col base (4 16-col tiles)

    // per-thread tile-load geometry (constant across chunks)
    int ar[4], ac[4];
    #pragma unroll
    for (int i2 = 0; i2 < 4; ++i2) {
        int fidx = tid + i2 * 256;
        ar[i2] = fidx >> 3;               // A: 8 float4 per 32-float row
        ac[i2] = (fidx & 7) * 4;
    }
    int bpr[2], bn0[2];
    #pragma unroll
    for (int i2 = 0; i2 < 2; ++i2) {
        int blk = tid + i2 * 256;         // 512 2x4 blocks in a 32x128 tile
        bpr[i2] = blk >> 5;               // k-pair row (0..15)
        bn0[i2] = (blk & 31) * 4;         // n base
    }

    v8f acc0[4], acc1[4];
    #pragma unroll
    for (int nt = 0; nt < 4; ++nt) { v8f z = {}; acc0[nt] = z; acc1[nt] = z; }

    // stage chunk 0 into registers
    vf4 ra[4], rb0[2], rb1[2];
    #pragma unroll
    for (int i2 = 0; i2 < 4; ++i2)
        ra[i2] = *(const vf4*)(Ab + (long)ar[i2] * K + ac[i2]);
    #pragma unroll
    for (int i2 = 0; i2 < 2; ++i2) {
        rb0[i2] = *(const vf4*)(Bb + (long)(2 * bpr[i2])     * DD + bn0[i2]);
        rb1[i2] = *(const vf4*)(Bb + (long)(2 * bpr[i2] + 1) * DD + bn0[i2]);
    }

    for (int kc = 0; kc < K; kc += KC) {
        __syncthreads();     // previous chunk's readers done
        // commit staged registers to LDS
        #pragma unroll
        for (int i2 = 0; i2 < 4; ++i2)
            *(vf4*)(&As[ar[i2] * ALDS + ac[i2]]) = ra[i2];
        #pragma unroll
        for (int i2 = 0; i2 < 2; ++i2) {
            vf4 lo, hi;
            lo[0] = rb0[i2][0]; lo[1] = rb1[i2][0]; lo[2] = rb0[i2][1]; lo[3] = rb1[i2][1];
            hi[0] = rb0[i2][2]; hi[1] = rb1[i2][2]; hi[2] = rb0[i2][3]; hi[3] = rb1[i2][3];
            float* dst = &Bs[bpr[i2] * PSTR + bn0[i2] * 2];
            *(vf4*)(dst)     = lo;
            *(vf4*)(dst + 4) = hi;
        }
        __syncthreads();

        // prefetch next chunk into registers (overlaps compute below)
        int kn = kc + KC;
        if (kn < K) {
            #pragma unroll
            for (int i2 = 0; i2 < 4; ++i2)
                ra[i2] = *(const vf4*)(Ab + (long)ar[i2] * K + kn + ac[i2]);
            #pragma unroll
            for (int i2 = 0; i2 < 2; ++i2) {
                rb0[i2] = *(const vf4*)(Bb + (long)(kn + 2 * bpr[i2])     * DD + bn0[i2]);
                rb1[i2] = *(const vf4*)(Bb + (long)(kn + 2 * bpr[i2] + 1) * DD + bn0[i2]);
            }
        }

        const float* arow0 = &As[(m0 + mrow) * ALDS + kh];
        const float* arow1 = arow0 + 16 * ALDS;
        const float* brow  = &Bs[(lane >> 4) * PSTR + (nh + mrow) * 2];
        #pragma unroll
        for (int kk = 0; kk < KC; kk += 4) {
            v2f a0 = *(const v2f*)(arow0 + kk);
            v2f a1 = *(const v2f*)(arow1 + kk);
            #pragma unroll
            for (int nt = 0; nt < 4; ++nt) {
                v2f b = *(const v2f*)(brow + (kk >> 1) * PSTR + nt * 32);
                acc0[nt] = __builtin_amdgcn_wmma_f32_16x16x4_f32(
                    false, a0, false, b, (short)0, acc0[nt], false, false);
                acc1[nt] = __builtin_amdgcn_wmma_f32_16x16x4_f32(
                    false, a1, false, b, (short)0, acc1[nt], false, false);
            }
        }
    }

    // C/D layout: VGPR r -> rows r (lanes 0-15) and r+8 (lanes 16-31)
    int rbase = (lane >> 4) * 8;
    #pragma unroll
    for (int nt = 0; nt < 4; ++nt) {
        #pragma unroll
        for (int r = 0; r < 8; ++r) {
            float v0 = acc0[nt][r];
            float v1 = acc1[nt][r];
            if (relu) { v0 = fmaxf(v0, 0.0f); v1 = fmaxf(v1, 0.0f); }
            int row0 = m0 + rbase + r;
            int col  = nh + nt * 16 + mrow;
            Cb[(long)row0 * DD + col]        = v0;
            Cb[(long)(row0 + 16) * DD + col] = v1;
        }
    }
}

extern "C" void kernel_launch(void* const* d_in, const int* in_sizes, int n_in,
                              void* d_out, int out_size, void* d_ws, size_t ws_size,
                              hipStream_t stream) {
    const int*   Nn    = (const int*)d_in[0];
    const float* E     = (const float*)d_in[1];
    const float* embed = (const float*)d_in[2];
    const float* W[3]  = { (const float*)d_in[3], (const float*)d_in[4], (const float*)d_in[5] };
    float* out = (float*)d_out;
    float* ws  = (float*)d_ws;

    float* En   = ws;                          // 96*512*512   = 25,165,824 f
    float* Hcur = En   + (long)NG * NN * NN;   // 96*512*128   =  6,291,456 f
    float* HW   = Hcur + (long)NG * NN * DD;   // 96*512*128   =  6,291,456 f
    float* dinv = HW   + (long)NG * NN * DD;   // 96*512       =     49,152 f

    // normalize adjacency
    rowsum_kernel<<<NG * NN / 8, 256, 0, stream>>>(E, dinv);
    en_kernel<<<(int)((long)NG * NN * NN / 4 / 256), 256, 0, stream>>>(E, dinv, En);
    // gather embeddings, broadcast across T
    gather_kernel<<<(int)((long)NG * NN * DD / 4 / 256), 256, 0, stream>>>(Nn, embed, Hcur);

    dim3 ggrid(NN / 128, NG);
    for (int l = 0; l < 3; ++l) {
        // HW = H @ W_l   (K = 128, B indexed by t = g % 3)
        gemm_wmma<<<ggrid, 256, 0, stream>>>(Hcur, (long)NN * DD, DD,
                                             W[l], (long)DD * DD, NT, HW, 0);
        // H' = relu(En @ HW)   (K = 512, B indexed by g)
        float* dst = (l == 2) ? out : Hcur;
        gemm_wmma<<<ggrid, 256, 0, stream>>>(En, (long)NN * NN, NN,
                                             HW, (long)NN * DD, 0, dst, 1);
    }
}